// slater_jastrow_68410239091339
// MI455X (gfx1250) — compile-verified
//
#include <hip/hip_runtime.h>
#include <stdint.h>
#include <math.h>

// Problem constants (match reference)
#define NORB   128          // orbitals per spin sector
#define NFERM  64           // fermions per sector -> 64x64 determinant
#define NSEC   2            // spin sectors
#define APAD   65           // complex elements per padded LDS row (bank-decorrelated)
#define TPB    256          // 8 wave32 waves per block
#define LN2_D  0.69314718055994530941723212145818

typedef unsigned int u32x4 __attribute__((ext_vector_type(4)));
typedef int          i32x4 __attribute__((ext_vector_type(4)));
typedef int          i32x8 __attribute__((ext_vector_type(8)));

#if defined(__has_builtin)
#  if __has_builtin(__builtin_amdgcn_tensor_load_to_lds)
#    define HAVE_TDM 1
#  endif
#endif
#ifndef HAVE_TDM
#  define HAVE_TDM 0
#endif

// ---- compile-probe diagnostics (visible in hipcc stderr) ----
#if HAVE_TDM
#  warning "CDNA5 probe: __builtin_amdgcn_tensor_load_to_lds AVAILABLE -> TDM gather path enabled"
#else
#  warning "CDNA5 probe: __builtin_amdgcn_tensor_load_to_lds NOT available -> global-load gather fallback"
#endif
#if __clang_major__ >= 23
#  warning "CDNA5 probe: clang >= 23 -> using 6-arg tensor_load_to_lds form"
#else
#  warning "CDNA5 probe: clang < 23 -> using 5-arg tensor_load_to_lds form"
#endif

struct Cplx { double re, im; };

// Dynamic LDS layout (~39 KB, < 64 KB dynamic-shared limit)
struct Smem {
  double2  buf[2][16 * APAD];   // TDM landing buffers (double-buffered 16-row chunks)
  double2  colv[NFERM];         // pivot-column value snapshot (pre-scale)
  double2  colbuf[NFERM];       // scaled L column broadcast
  double2  rowbuf[NFERM];       // U row broadcast
  double   smag[NFERM];         // pivot magnitudes
  double   rmag[16];            // argmax partials
  int      ridx[16];
  int      perm[NFERM];         // chosen physical pivot row per step
  int      spiv;
  int      scan[2][NORB];       // occupancy prefix-scan ping-pong
  int      occ[NFERM];          // occupied orbital indices, ascending
  unsigned occ16[NFERM / 2];    // same, packed 2x16-bit for TDM gather descriptors
};

#if HAVE_TDM
// Issue one TDM gather-mode load: 16 rows of M (selected by 16-bit indices)
// into S->buf[d&1], padded to APAD complex per row.
__device__ __forceinline__ void tdm_issue(Smem* S, uint64_t mga, int d) {
  u32x4 g0;
  g0[0] = 0x80000001u;                                   // count=1, gather_mode=1, 16-bit idx
  g0[1] = (uint32_t)(uintptr_t)(&S->buf[d & 1][0]);      // LDS destination (bytes)
  g0[2] = (uint32_t)mga;                                 // global addr lo (tile == tensor start)
  g0[3] = ((uint32_t)(mga >> 32) & 0x01FFFFFFu) | (2u << 30);  // addr hi + type=2 (image)
  i32x8 g1;
  // data_size=3 (8B), pad_enable, pad_interval=7 (256 dw), pad_amount code 3 (4 dw -> 65-complex stride)
  g1[0] = (3 << 16) | (1 << 20) | (7 << 22) | (3 << 25);
  g1[1] = (128 << 16);   // tensor_dim0 = 128 (8B units per 64-complex row), low 16 bits
  g1[2] = (128 << 16);   // tensor_dim0 hi = 0 | tensor_dim1 = 128 rows (low 16)
  g1[3] = (128 << 16);   // tensor_dim1 hi = 0 | tile_dim0 = 128
  g1[4] = 16;            // tile_dim1 = 16 valid gather indices
  g1[5] = 128;           // tensor_dim0_stride = 128 (8B units)
  g1[6] = 0;
  g1[7] = 0;
  i32x4 g2, g3;
#pragma unroll
  for (int q = 0; q < 4; ++q) {
    g2[q] = __builtin_amdgcn_readfirstlane((int)S->occ16[d * 8 + q]);
    g3[q] = __builtin_amdgcn_readfirstlane((int)S->occ16[d * 8 + 4 + q]);
  }
#if __clang_major__ >= 23
  i32x8 g4 = {0, 0, 0, 0, 0, 0, 0, 0};
  __builtin_amdgcn_tensor_load_to_lds(g0, g1, g2, g3, g4, 0);
#else
  __builtin_amdgcn_tensor_load_to_lds(g0, g1, g2, g3, 0);
#endif
}
#endif

__global__ void __launch_bounds__(TPB)
slater_logdet_kernel(const int* __restrict__ nocc,
                     const double* __restrict__ Mraw,
                     double2* __restrict__ out)
{
  extern __shared__ unsigned char smem_raw[];
  Smem* S = (Smem*)smem_raw;
  const double2* Mc = (const double2*)Mraw;

  const int tid = threadIdx.x;
  const int tx  = tid & 15;       // column group owner
  const int ty  = tid >> 4;       // row group owner
  const int b   = blockIdx.x;

  double acc_re = 0.0, acc_im = 0.0;  // thread 0 accumulates across sectors
  Cplx a[4][4];                       // register-resident 4x4 complex tile of A

  for (int s = 0; s < NSEC; ++s) {
    // ---------- stable prefix-scan of occupation flags (ascending indices) ----------
    int flag = 0;
    if (tid < NORB) {
      flag = (nocc[(size_t)b * (NSEC * NORB) + (size_t)s * NORB + tid] != 0) ? 1 : 0;
      S->scan[0][tid] = flag;
    }
    __syncthreads();
    int src = 0;
    for (int st = 1; st < NORB; st <<= 1) {
      if (tid < NORB) {
        int v = S->scan[src][tid];
        if (tid >= st) v += S->scan[src][tid - st];
        S->scan[src ^ 1][tid] = v;
      }
      __syncthreads();
      src ^= 1;
    }
    if (tid < NORB && flag) S->occ[S->scan[src][tid] - 1] = tid;
    __syncthreads();

    // ---------- gather A = M[occ] into registers ----------
#if HAVE_TDM
    if (tid < NFERM / 2)
      S->occ16[tid] = (unsigned)S->occ[2 * tid] | ((unsigned)S->occ[2 * tid + 1] << 16);
    __syncthreads();
    {
      const uint64_t mga = (uint64_t)(uintptr_t)Mraw;
      if (tid < 32) tdm_issue(S, mga, 0);     // wave 0 drives the TDM, EXEC all-ones
#pragma unroll
      for (int d = 0; d < 4; ++d) {
        if (tid < 32) {
          if (d < 3) {
            tdm_issue(S, mga, d + 1);               // keep next chunk in flight
            __builtin_amdgcn_s_wait_tensorcnt(1);   // wait for chunk d only
          } else {
            __builtin_amdgcn_s_wait_tensorcnt(0);
          }
        }
        __syncthreads();
        const double2* bb = S->buf[d & 1];
#pragma unroll
        for (int n = 0; n < 4; ++n) {
          double2 t = bb[ty * APAD + tx + 16 * n];
          a[d][n].re = t.x; a[d][n].im = t.y;
        }
        __syncthreads();
      }
    }
#else
    {
      int rows[4];
#pragma unroll
      for (int m = 0; m < 4; ++m) rows[m] = S->occ[ty + 16 * m];
#pragma unroll
      for (int m = 0; m < 4; ++m)
#pragma unroll
        for (int n = 0; n < 4; ++n) {
          double2 t = Mc[(size_t)rows[m] * NFERM + tx + 16 * n];  // coalesced b128 loads
          a[m][n].re = t.x; a[m][n].im = t.y;
        }
      __syncthreads();
    }
#endif

    // ---------- register-resident complex LU, pivoting WITHOUT row swaps ----------
    // Per-thread elimination mask over its 4 owned rows; permutation parity fixed at end.
    unsigned elim = 0u;
    double P_re = 1.0, P_im = 0.0;  // renormalized product of pivots (uniform on all lanes)
    int totE = 0;                   // accumulated power-of-two exponent

#pragma unroll 1
    for (int k = 0; k < NFERM; ++k) {
      const int k15 = k & 15;
      const int km  = k >> 4;

      // (a) column owners publish pivot-column values + magnitudes of live rows
      if (tx == k15) {
#pragma unroll
        for (int m = 0; m < 4; ++m) {
          int i = ty + 16 * m;
          Cplx v = a[m][km];
          S->colv[i] = make_double2(v.re, v.im);
          S->smag[i] = ((elim >> m) & 1u) ? -1.0 : (v.re * v.re + v.im * v.im);
        }
      }
      __syncthreads();

      // (b) flat argmax: 64 -> 16 -> 1
      if (tid < 16) {
        int base = tid << 2;
        double best = S->smag[base]; int bi = base;
#pragma unroll
        for (int q = 1; q < 4; ++q) {
          double v = S->smag[base + q];
          if (v > best) { best = v; bi = base + q; }
        }
        S->rmag[tid] = best; S->ridx[tid] = bi;
      }
      __syncthreads();
      if (tid == 0) {
        double best = S->rmag[0]; int bi = S->ridx[0];
#pragma unroll
        for (int q = 1; q < 16; ++q) {
          double v = S->rmag[q];
          if (v > best) { best = v; bi = S->ridx[q]; }
        }
        S->spiv = bi;
        S->perm[k] = bi;
      }
      __syncthreads();
      const int p   = S->spiv;
      const int p15 = p & 15;
      const int pm  = p >> 4;

      // mark pivot row eliminated (consistent across all 16 owners of that ty group)
      if (ty == p15) elim |= (1u << pm);

      // pivot broadcast; all lanes redundantly (uniformly) form 1/pivot and update P
      double2 pvv = S->colv[p];
      {
        double d2 = pvv.x * pvv.x + pvv.y * pvv.y;
        double nr = P_re * pvv.x - P_im * pvv.y;
        double ni = P_re * pvv.y + P_im * pvv.x;
        double m2 = nr * nr + ni * ni;
        int h = ilogb(m2) >> 1;          // exact power-of-two renormalization
        P_re = ldexp(nr, -h);
        P_im = ldexp(ni, -h);
        totE += h;
      }
      double rd  = 1.0 / (pvv.x * pvv.x + pvv.y * pvv.y);
      double ivr =  pvv.x * rd;
      double ivi = -pvv.y * rd;

      // (e) publish scaled L column (live rows) and U row (pivot row, j > k)
      if (tx == k15) {
#pragma unroll
        for (int m = 0; m < 4; ++m) {
          if (!((elim >> m) & 1u)) {
            int i = ty + 16 * m;
            double lr = a[m][km].re * ivr - a[m][km].im * ivi;
            double li = a[m][km].re * ivi + a[m][km].im * ivr;
            S->colbuf[i] = make_double2(lr, li);
          }
        }
      }
      if (ty == p15) {
#pragma unroll
        for (int n = 0; n < 4; ++n) {
          int j = tx + 16 * n;
          if (j > k) S->rowbuf[j] = make_double2(a[pm][n].re, a[pm][n].im);
        }
      }
      __syncthreads();

      // (f) rank-1 trailing update on live rows, entirely in registers
#pragma unroll
      for (int m = 0; m < 4; ++m) {
        if (!((elim >> m) & 1u)) {
          double2 l = S->colbuf[ty + 16 * m];
#pragma unroll
          for (int n = 0; n < 4; ++n) {
            int j = tx + 16 * n;
            if (j > k) {
              double2 u = S->rowbuf[j];
              a[m][n].re -= l.x * u.x - l.y * u.y;
              a[m][n].im -= l.x * u.y + l.y * u.x;
            }
          }
        }
      }
      // no barrier needed: next step's (a) touches disjoint LDS, own registers
    } // k

    // ---------- sector wrap-up (thread 0): permutation parity + final log/atan2 ----------
    __syncthreads();   // ensure perm[] fully written / LDS reusable
    if (tid == 0) {
      unsigned long long visited = 0ull;
      int transpositions = 0;
      for (int k0 = 0; k0 < NFERM; ++k0) {
        if ((visited >> k0) & 1ull) continue;
        int len = 0, j = k0;
        while (!((visited >> j) & 1ull)) {
          visited |= (1ull << j);
          j = S->perm[j];
          ++len;
        }
        transpositions += len - 1;
      }
      double pr = P_re, pi = P_im;
      if (transpositions & 1) { pr = -pr; pi = -pi; }
      double m2 = pr * pr + pi * pi;
      acc_re += 0.5 * log(m2) + LN2_D * (double)totE;  // log|det|
      acc_im += atan2(pi, pr);                          // principal arg(det)
    }
    __syncthreads();  // LDS reused by next sector
  } // sectors

  if (tid == 0) out[b] = make_double2(acc_re, acc_im);
}

extern "C" void kernel_launch(void* const* d_in, const int* in_sizes, int n_in,
                              void* d_out, int out_size, void* d_ws, size_t ws_size,
                              hipStream_t stream) {
  (void)n_in; (void)out_size; (void)d_ws; (void)ws_size;
  const int*    nocc = (const int*)d_in[0];      // [B, 2*128] int32 occupation flags
  const double* M    = (const double*)d_in[1];   // [128, 64] complex128 (interleaved)
  double2*      out  = (double2*)d_out;          // [B] complex128

  int batch = in_sizes[0] / (NSEC * NORB);
  if (batch <= 0) return;

  dim3 grid((unsigned)batch), block(TPB);
  size_t smem = sizeof(Smem);
  hipLaunchKernelGGL(slater_logdet_kernel, grid, block, smem, stream, nocc, M, out);
}